// Gate_75496935129437
// MI455X (gfx1250) — compile-verified
//
#include <hip/hip_runtime.h>

// MoE gate: logits = x @ W^T, fp32 softmax over 4 experts, +bias for selection,
// top-2 indices, weights gathered from pre-bias scores.
// Memory-bound: streams 256 MB of x once. fp32 WMMA (16x16x4) keeps routing
// precision identical to the fp32 reference.

#define TOKENS 131072
#define DIM 512
#define NEXP 4
#define TOK_PER_BLOCK 128   // 8 waves * 16 tokens

typedef __attribute__((ext_vector_type(2))) float v2f;
typedef __attribute__((ext_vector_type(8))) float v8f;

__global__ __launch_bounds__(256) void moe_gate_wmma_kernel(
    const float* __restrict__ x,       // [TOKENS, DIM]
    const float* __restrict__ weight,  // [NEXP, DIM]
    const float* __restrict__ bias,    // [NEXP]
    float* __restrict__ wout,          // [TOKENS, 2]
    int* __restrict__ iout)            // [TOKENS, 2]
{
  __shared__ float s_logits[TOK_PER_BLOCK * NEXP];

  const int tid  = threadIdx.x;
  const int wave = tid >> 5;
  const int lane = tid & 31;
  const int l15  = lane & 15;   // A: token row within tile; B: expert column
  const int half = lane >> 4;   // selects K-subrange of the fragment

  const int t0 = blockIdx.x * TOK_PER_BLOCK + wave * 16;

  // A fragment: lane holds x[t0+l15][k0 + half*2 .. +1]  (8B aligned pair)
  const float* abase = x + (size_t)(t0 + l15) * DIM + half * 2;
  // B fragment: lane holds W[l15][k0 + half*2 .. +1]; columns >=4 alias rows 0-3
  // (their products land in D columns 4..15 which are never read).
  const float* bbase = weight + (size_t)(l15 & 3) * DIM + half * 2;

  v8f c = {};
#pragma unroll 8
  for (int k0 = 0; k0 < DIM; k0 += 4) {
    v2f a = *(const v2f*)(abase + k0);
    v2f b = *(const v2f*)(bbase + k0);
    // 8 args: (neg_a, A, neg_b, B, c_mod, C, reuse_a, reuse_b)
    c = __builtin_amdgcn_wmma_f32_16x16x4_f32(
        false, a, false, b, (short)0, c, false, false);
  }

  // D layout: VGPR g -> (M=g, N=lane) for lanes 0-15, (M=g+8, N=lane-16) for 16-31.
  // Real logits live in columns (l15) 0..3.
  if (l15 < NEXP) {
    const int tl = wave * 16 + half * 8;
#pragma unroll
    for (int g = 0; g < 8; ++g)
      s_logits[(tl + g) * NEXP + l15] = c[g];
  }
  __syncthreads();

  if (tid < TOK_PER_BLOCK) {
    const int t = blockIdx.x * TOK_PER_BLOCK + tid;
    const float l0 = s_logits[tid * NEXP + 0];
    const float l1 = s_logits[tid * NEXP + 1];
    const float l2 = s_logits[tid * NEXP + 2];
    const float l3 = s_logits[tid * NEXP + 3];

    // fp32 softmax with max subtraction (matches jax.nn.softmax numerics)
    const float m = fmaxf(fmaxf(l0, l1), fmaxf(l2, l3));
    const float e0 = __expf(l0 - m);
    const float e1 = __expf(l1 - m);
    const float e2 = __expf(l2 - m);
    const float e3 = __expf(l3 - m);
    const float inv = 1.0f / (e0 + e1 + e2 + e3);
    const float s0 = e0 * inv, s1 = e1 * inv, s2 = e2 * inv, s3 = e3 * inv;

    // bias only affects selection
    const float c0 = s0 + bias[0];
    const float c1 = s1 + bias[1];
    const float c2 = s2 + bias[2];
    const float c3 = s3 + bias[3];

    // top-1: strict > keeps first occurrence (jax top_k tie-break = lower index)
    int i0 = 0; float b0 = c0;
    if (c1 > b0) { b0 = c1; i0 = 1; }
    if (c2 > b0) { b0 = c2; i0 = 2; }
    if (c3 > b0) { b0 = c3; i0 = 3; }
    // top-2: first max excluding i0
    int i1 = -1; float b1 = -3.402823466e38f;
    if (i0 != 0 && c0 > b1) { b1 = c0; i1 = 0; }
    if (i0 != 1 && c1 > b1) { b1 = c1; i1 = 1; }
    if (i0 != 2 && c2 > b1) { b1 = c2; i1 = 2; }
    if (i0 != 3 && c3 > b1) { b1 = c3; i1 = 3; }

    // gather weights from PRE-bias scores (ROUTE_SCALE == 1.0)
    const float w0 = (i0 == 0) ? s0 : (i0 == 1) ? s1 : (i0 == 2) ? s2 : s3;
    const float w1 = (i1 == 0) ? s0 : (i1 == 1) ? s1 : (i1 == 2) ? s2 : s3;

    wout[(size_t)t * 2 + 0] = w0;
    wout[(size_t)t * 2 + 1] = w1;
    iout[(size_t)t * 2 + 0] = i0;
    iout[(size_t)t * 2 + 1] = i1;
  }
}

extern "C" void kernel_launch(void* const* d_in, const int* in_sizes, int n_in,
                              void* d_out, int out_size, void* d_ws, size_t ws_size,
                              hipStream_t stream) {
  (void)in_sizes; (void)n_in; (void)out_size; (void)d_ws; (void)ws_size;
  const float* x      = (const float*)d_in[0];
  const float* weight = (const float*)d_in[1];
  const float* bias   = (const float*)d_in[2];

  float* wout = (float*)d_out;                       // [TOKENS,2] fp32
  int*   iout = (int*)(wout + (size_t)TOKENS * 2);   // [TOKENS,2] int32 (concat)

  dim3 block(256);
  dim3 grid(TOKENS / TOK_PER_BLOCK);                 // 1024 blocks
  moe_gate_wmma_kernel<<<grid, block, 0, stream>>>(x, weight, bias, wout, iout);
}